// Refine_Attn_20512763805917
// MI455X (gfx1250) — compile-verified
//
#include <hip/hip_runtime.h>
#include <math.h>

typedef __attribute__((ext_vector_type(2))) float v2f;
typedef __attribute__((ext_vector_type(8))) float v8f;

#define KS 5
#define NEIGH 25

// ---------------------------------------------------------------------------
// GEMM: Y[M,N] = X[M,K] @ W[K,N] + bias[N], fp32 via V_WMMA_F32_16X16X4_F32.
// One wave computes one 16x16 tile; 4 waves per block tile along M.
// M % 64 == 0, N % 16 == 0, K % 4 == 0 for all call sites here.
// ---------------------------------------------------------------------------
__global__ __launch_bounds__(128) void gemm_f32_wmma(
    const float* __restrict__ X, const float* __restrict__ Wm,
    const float* __restrict__ bias, float* __restrict__ Y,
    int M, int N, int K)
{
    const int lane = threadIdx.x & 31;
    const int wave = threadIdx.x >> 5;
    const int m0 = blockIdx.y * 64 + wave * 16;
    const int n0 = blockIdx.x * 16;
    const int half = lane >> 4;     // 0: K pair {k0,k0+1}, 1: {k0+2,k0+3}
    const int lm   = lane & 15;     // row (A) / col (B,C,D) within tile

    v8f acc = {};
    const float* arow = X + (size_t)(m0 + lm) * K;
    for (int k0 = 0; k0 < K; k0 += 4) {
        const int ka = k0 + 2 * half;
        v2f a, b;
        a.x = arow[ka];
        a.y = arow[ka + 1];
        b.x = Wm[(size_t)ka       * N + n0 + lm];
        b.y = Wm[(size_t)(ka + 1) * N + n0 + lm];
        acc = __builtin_amdgcn_wmma_f32_16x16x4_f32(
            /*neg_a=*/false, a, /*neg_b=*/false, b,
            /*c_mod=*/(short)0, acc, /*reuse_a=*/false, /*reuse_b=*/false);
    }

    const float bv = bias ? bias[n0 + lm] : 0.0f;
    #pragma unroll
    for (int r = 0; r < 8; ++r) {
        const int row = m0 + r + 8 * half;      // C/D layout: lanes16-31 hold M=r+8
        Y[(size_t)row * N + n0 + lm] = acc[r] + bv;
    }
}

// ---------------------------------------------------------------------------
// Neighborhood attention (5x5 window, clamped to image), one wave32 per pixel.
// qkv layout: [HW][3C] = q|k|v.  out: [HW][C] (pre-projection).
// ---------------------------------------------------------------------------
__global__ __launch_bounds__(256) void na2d_attn(
    const float* __restrict__ qkv, const float* __restrict__ rpb,
    float* __restrict__ out, int H, int W, int C)
{
    const int lane = threadIdx.x & 31;
    const int wave = threadIdx.x >> 5;
    const int p = blockIdx.x * 8 + wave;
    if (p >= H * W) return;

    const int h = p / W, w = p % W;
    const int sh = min(max(h - KS / 2, 0), H - KS);
    const int sw = min(max(w - KS / 2, 0), W - KS);
    const float scale = rsqrtf((float)C);

    const float* q = qkv + (size_t)p * 3 * C;

    // --- logits on lanes 0..24 ---
    float logit = -INFINITY;
    if (lane < NEIGH) {
        const int i = lane / KS, j = lane % KS;
        const int ih = sh + i, iw = sw + j;
        const float* kk = qkv + (size_t)(ih * W + iw) * 3 * C + C;
        float d = 0.f;
        for (int c = 0; c < C; ++c) d += q[c] * kk[c];
        const int bi = ih - h + (KS - 1);
        const int bj = iw - w + (KS - 1);
        logit = d * scale + rpb[bi * (2 * KS - 1) + bj];
    }

    // --- wave32 softmax over the 25 logits ---
    float mx = logit;
    for (int m = 16; m >= 1; m >>= 1) mx = fmaxf(mx, __shfl_xor(mx, m, 32));
    float e = (lane < NEIGH) ? __expf(logit - mx) : 0.f;
    float s = e;
    for (int m = 16; m >= 1; m >>= 1) s += __shfl_xor(s, m, 32);
    const float prob = e / s;

    // --- weighted sum of V: lane handles channels lane, lane+32, ... ---
    const int nc = C >> 5;              // C in {32,64,128}
    float acc[4] = {0.f, 0.f, 0.f, 0.f};
    for (int j = 0; j < NEIGH; ++j) {
        const float aj = __shfl(prob, j, 32);
        const int pn = (sh + j / KS) * W + (sw + j % KS);
        const float* v = qkv + (size_t)pn * 3 * C + 2 * C;
        #pragma unroll 4
        for (int t = 0; t < nc; ++t) acc[t] += aj * v[lane + (t << 5)];
    }
    for (int t = 0; t < nc; ++t)
        out[(size_t)p * C + lane + (t << 5)] = acc[t];
}

// ---------------------------------------------------------------------------
// Bilinear 2x upsample (align_corners-style pos = y*(n-1)/(2n-1)), HWC layout.
// ---------------------------------------------------------------------------
__global__ void up2x_bilinear(const float* __restrict__ in, float* __restrict__ out,
                              int H, int W, int C)
{
    const int H2 = 2 * H, W2 = 2 * W;
    const size_t total = (size_t)H2 * W2 * C;
    const size_t idx = (size_t)blockIdx.x * blockDim.x + threadIdx.x;
    if (idx >= total) return;

    const int c = (int)(idx % C);
    const int x = (int)((idx / C) % W2);
    const int y = (int)(idx / ((size_t)C * W2));

    const float py = y * (float)(H - 1) / (float)(H2 - 1);
    const float px = x * (float)(W - 1) / (float)(W2 - 1);
    const int i0 = (int)floorf(py); const int i1 = min(i0 + 1, H - 1);
    const int j0 = (int)floorf(px); const int j1 = min(j0 + 1, W - 1);
    const float wy = py - (float)i0;
    const float wx = px - (float)j0;

    const float* r0 = in + (size_t)i0 * W * C;
    const float* r1 = in + (size_t)i1 * W * C;
    const float v00 = r0[(size_t)j0 * C + c], v01 = r0[(size_t)j1 * C + c];
    const float v10 = r1[(size_t)j0 * C + c], v11 = r1[(size_t)j1 * C + c];
    const float top = v00 * (1.f - wx) + v01 * wx;
    const float bot = v10 * (1.f - wx) + v11 * wx;
    out[idx] = top * (1.f - wy) + bot * wy;
}

// ---------------------------------------------------------------------------
// Layout transposes: CHW <-> HWC
// ---------------------------------------------------------------------------
__global__ void chw_to_hwc(const float* __restrict__ in, float* __restrict__ out,
                           int C, int H, int W)
{
    const size_t total = (size_t)C * H * W;
    const size_t idx = (size_t)blockIdx.x * blockDim.x + threadIdx.x;
    if (idx >= total) return;
    const int c = (int)(idx % C);
    const size_t hw = idx / C;
    out[idx] = in[(size_t)c * H * W + hw];
}

__global__ void hwc_to_chw(const float* __restrict__ in, float* __restrict__ out,
                           int C, int H, int W)
{
    const size_t total = (size_t)C * H * W;
    const size_t idx = (size_t)blockIdx.x * blockDim.x + threadIdx.x;
    if (idx >= total) return;
    const size_t hw = idx % ((size_t)H * W);
    const int c = (int)(idx / ((size_t)H * W));
    out[idx] = in[hw * C + c];
}

// ---------------------------------------------------------------------------
// Host orchestration
// ---------------------------------------------------------------------------
extern "C" void kernel_launch(void* const* d_in, const int* in_sizes, int n_in,
                              void* d_out, int out_size, void* d_ws, size_t ws_size,
                              hipStream_t stream)
{
    (void)in_sizes; (void)n_in; (void)out_size; (void)ws_size;

    const float* attn = (const float*)d_in[0];
    const float* w_qkv[3]  = {(const float*)d_in[1],  (const float*)d_in[6],  (const float*)d_in[11]};
    const float* b_qkv[3]  = {(const float*)d_in[2],  (const float*)d_in[7],  (const float*)d_in[12]};
    const float* w_proj[3] = {(const float*)d_in[3],  (const float*)d_in[8],  (const float*)d_in[13]};
    const float* b_proj[3] = {(const float*)d_in[4],  (const float*)d_in[9],  (const float*)d_in[14]};
    const float* rpb[3]    = {(const float*)d_in[5],  (const float*)d_in[10], (const float*)d_in[15]};
    const float* w_up[2]   = {(const float*)d_in[16], (const float*)d_in[18]};
    const float* b_up[2]   = {(const float*)d_in[17], (const float*)d_in[19]};

    // Workspace layout (floats): X (2.36M) | QKV (7.08M) | T (2.36M)
    float* X   = (float*)d_ws;
    float* QKV = X + (size_t)2359296;
    float* T   = QKV + (size_t)7077888;

    const int Hs[3] = {24, 48, 96};
    const int Ws[3] = {48, 96, 192};
    const int Cs[3] = {32, 64, 128};

    auto gemm = [&](const float* A, const float* Wm, const float* bias, float* Y,
                    int M, int N, int K) {
        dim3 grid(N / 16, M / 64);
        gemm_f32_wmma<<<grid, 128, 0, stream>>>(A, Wm, bias, Y, M, N, K);
    };

    // input transpose: (1,C,H,W) -> (H,W,C)
    {
        const size_t total = (size_t)Cs[0] * Hs[0] * Ws[0];
        chw_to_hwc<<<(unsigned)((total + 255) / 256), 256, 0, stream>>>(
            attn, X, Cs[0], Hs[0], Ws[0]);
    }

    for (int s = 0; s < 3; ++s) {
        const int H = Hs[s], W = Ws[s], C = Cs[s];
        const int HW = H * W;

        // qkv = X @ w_qkv + b_qkv         [HW, 3C]
        gemm(X, w_qkv[s], b_qkv[s], QKV, HW, 3 * C, C);

        // neighborhood attention -> T     [HW, C]
        na2d_attn<<<(HW + 7) / 8, 256, 0, stream>>>(QKV, rpb[s], T, H, W, C);

        // X = T @ w_proj + b_proj         [HW, C]
        gemm(T, w_proj[s], b_proj[s], X, HW, C, C);

        if (s < 2) {
            // T = up2x(X)                 [2H, 2W, C]
            const size_t total = (size_t)4 * HW * C;
            up2x_bilinear<<<(unsigned)((total + 255) / 256), 256, 0, stream>>>(
                X, T, H, W, C);
            // X = T @ w_up + b_up         [4HW, 2C]
            gemm(T, w_up[s], b_up[s], X, 4 * HW, 2 * C, C);
        }
    }

    // final transpose: (H,W,C) -> (1,C,H,W)
    {
        const size_t total = (size_t)Cs[2] * Hs[2] * Ws[2];
        hwc_to_chw<<<(unsigned)((total + 255) / 256), 256, 0, stream>>>(
            X, (float*)d_out, Cs[2], Hs[2], Ws[2]);
    }
}